// VMapCompatibleGCN_31688268710058
// MI455X (gfx1250) — compile-verified
//
#include <hip/hip_runtime.h>
#include <hip/hip_bf16.h>

#define EPS 1e-6f
#define CH 128  // Cin == Cout == 128

typedef float v2f __attribute__((ext_vector_type(2)));
typedef float v8f __attribute__((ext_vector_type(8)));

// ---------------------------------------------------------------------------
// Kernel 1: zero workspace (agg + degree arrays)
// ---------------------------------------------------------------------------
__global__ void gcn_zero_kernel(float* __restrict__ p, size_t n) {
    size_t i = (size_t)blockIdx.x * blockDim.x + threadIdx.x;
    if (i < n) p[i] = 0.0f;
}

// ---------------------------------------------------------------------------
// Kernel 2: degree counts via global float atomics
// ---------------------------------------------------------------------------
__global__ void gcn_degree_kernel(const int* __restrict__ src,
                                  const int* __restrict__ dst,
                                  float* __restrict__ src_deg,
                                  float* __restrict__ dst_deg,
                                  int E) {
    int e = blockIdx.x * blockDim.x + threadIdx.x;
    if (e >= E) return;
    atomicAdd(&src_deg[src[e]], 1.0f);
    atomicAdd(&dst_deg[dst[e]], 1.0f);
}

// ---------------------------------------------------------------------------
// Kernel 3: edge scatter. One wave32 per edge; each lane handles 4 channels
// (32 lanes x float4 = 128 channels). The per-edge index/degree loads are
// wave-uniform -> force them onto the scalar path (SMEM/constant cache) via
// readfirstlane so vector-memory bandwidth is reserved for the x gather and
// the f32 atomics (the real bottleneck; both L2-resident working sets).
// ---------------------------------------------------------------------------
__global__ void gcn_scatter_kernel(const float* __restrict__ x,
                                   const int* __restrict__ src,
                                   const int* __restrict__ dst,
                                   const float* __restrict__ src_deg,
                                   const float* __restrict__ dst_deg,
                                   float* __restrict__ agg,
                                   int E) {
    int gtid = blockIdx.x * blockDim.x + threadIdx.x;
    int edge = __builtin_amdgcn_readfirstlane(gtid >> 5);  // wave-uniform SGPR
    int lane = threadIdx.x & 31;
    if (edge >= E) return;

    const int s = src[edge];   // scalar loads (uniform index)
    const int d = dst[edge];
    const float norm = rsqrtf(src_deg[s] + EPS) * rsqrtf(dst_deg[d] + EPS);

    const float4 xv = ((const float4*)(x + (size_t)s * CH))[lane];
    float* outp = agg + (size_t)d * CH + lane * 4;
    atomicAdd(outp + 0, xv.x * norm);
    atomicAdd(outp + 1, xv.y * norm);
    atomicAdd(outp + 2, xv.z * norm);
    atomicAdd(outp + 3, xv.w * norm);
}

// ---------------------------------------------------------------------------
// Kernel 4: out = agg[Nx128] @ W[128x128] + bias via V_WMMA_F32_16X16X4_F32.
// One wave per 16x16 output tile; 8 waves/block cover all 8 N-tiles of one
// M-tile (shared A fragments hit in L0 across the 8 waves).
//
// Layouts (wave32):
//   A 16x4  f32 (2 VGPR): lane l -> M = l%16, K = v + 2*(l/16)
//   B 4x16  f32 (2 VGPR): lane l -> N = l%16, K = v + 2*(l/16)
//   C/D 16x16 f32 (8 VGPR): lane l, vgpr v -> M = v + 8*(l/16), N = l%16
// ---------------------------------------------------------------------------
__global__ void gcn_gemm_wmma_kernel(const float* __restrict__ agg,
                                     const float* __restrict__ W,
                                     const float* __restrict__ bias,
                                     float* __restrict__ out,
                                     int N) {
    const int lane    = threadIdx.x & 31;
    const int wave    = threadIdx.x >> 5;      // 0..7  -> N tile
    const int tileM   = blockIdx.x;
    const int halfsel = lane >> 4;             // 0|1
    const int lmod    = lane & 15;

    const int row0 = tileM * 16;
    const int col0 = wave * 16;

    int aRow = row0 + lmod;
    if (aRow >= N) aRow = N - 1;               // clamp: keep EXEC all-1s for WMMA
    const int bCol = col0 + lmod;

    const float* aBase = agg + (size_t)aRow * CH;

    v8f c = {};
#pragma unroll 8
    for (int k = 0; k < CH; k += 4) {
        const int ka = k + 2 * halfsel;        // even -> 8B aligned
        v2f a, b;
        {
            const float2 av = *(const float2*)(aBase + ka);
            a.x = av.x; a.y = av.y;
        }
        b.x = W[(size_t)ka * CH + bCol];
        b.y = W[(size_t)(ka + 1) * CH + bCol];
        // D = A*B + C : v_wmma_f32_16x16x4_f32
        c = __builtin_amdgcn_wmma_f32_16x16x4_f32(
                /*neg_a=*/false, a, /*neg_b=*/false, b,
                /*c_mod=*/(short)0, c, /*reuse_a=*/false, /*reuse_b=*/false);
    }

    const float bv = bias[bCol];
    // Block-uniform bound check -> scalar branch; hot path has no exec-mask
    // manipulation, just 8 straight stores at stride CH.
    if (row0 + 16 <= N) {
        float* outp = out + ((size_t)row0 + 8 * halfsel) * CH + bCol;
#pragma unroll
        for (int v = 0; v < 8; ++v) {
            outp[(size_t)v * CH] = c[v] + bv;
        }
    } else {
#pragma unroll
        for (int v = 0; v < 8; ++v) {
            const int row = row0 + v + 8 * halfsel;
            if (row < N) out[(size_t)row * CH + bCol] = c[v] + bv;
        }
    }
}

// ---------------------------------------------------------------------------
// Launch: inputs = { x[N*128] f32, edge_index[2*E] i32, W[128*128] f32,
//                    bias[128] f32 }, output = out[N*128] f32.
// ws layout: agg[N*128] | src_deg[N] | dst_deg[N]   (floats)
// ---------------------------------------------------------------------------
extern "C" void kernel_launch(void* const* d_in, const int* in_sizes, int n_in,
                              void* d_out, int out_size, void* d_ws, size_t ws_size,
                              hipStream_t stream) {
    const float* x    = (const float*)d_in[0];
    const int*   edge = (const int*)d_in[1];
    const float* W    = (const float*)d_in[2];
    const float* bias = (const float*)d_in[3];
    float*       out  = (float*)d_out;

    const int N = in_sizes[0] / CH;
    const int E = in_sizes[1] / 2;
    const int* src = edge;
    const int* dst = edge + E;

    float* agg     = (float*)d_ws;
    float* src_deg = agg + (size_t)N * CH;
    float* dst_deg = src_deg + N;

    // 1) zero agg + degrees
    {
        size_t ztotal = (size_t)N * CH + 2 * (size_t)N;
        int blocks = (int)((ztotal + 255) / 256);
        gcn_zero_kernel<<<blocks, 256, 0, stream>>>((float*)d_ws, ztotal);
    }
    // 2) degrees
    gcn_degree_kernel<<<(E + 255) / 256, 256, 0, stream>>>(src, dst, src_deg, dst_deg, E);
    // 3) edge scatter: 1 wave per edge, 8 waves (256 threads) per block
    gcn_scatter_kernel<<<(E + 7) / 8, 256, 0, stream>>>(x, src, dst, src_deg, dst_deg, agg, E);
    // 4) WMMA GEMM + bias: one block per 16-row tile, 8 waves cover 128 cols
    gcn_gemm_wmma_kernel<<<(N + 15) / 16, 256, 0, stream>>>(agg, W, bias, out, N);
}